// GfusedmaxN_73521250173288
// MI455X (gfx1250) — compile-verified
//
#include <hip/hip_runtime.h>
#include <hip/hip_bf16.h>

typedef float v2f __attribute__((ext_vector_type(2)));
typedef float v8f __attribute__((ext_vector_type(8)));

#define LAMF      1.0f
#define TV_ITERS  20
#define NMAX_PAD  1536
#define SORTN     2048
#define PM_TPB    512

#if __has_builtin(__builtin_amdgcn_global_load_async_to_lds_b32)
#define HAVE_ASYNC_LDS 1
typedef __attribute__((address_space(1))) int gint_as1;   // global (AS1)
typedef __attribute__((address_space(3))) int lint_as3;   // LDS (AS3)
#endif

// ---------------- Phase 0: setup kernels ----------------

__global__ void k_zero(float* __restrict__ u, int* __restrict__ deg,
                       int* __restrict__ maxdeg, int E, int N) {
    int i = blockIdx.x * blockDim.x + threadIdx.x;
    if (i < E) u[i] = 0.0f;
    if (i < N) deg[i] = 0;
    if (i == 0) *maxdeg = 0;
}

__global__ void k_deg(const int* __restrict__ edges, int* __restrict__ deg, int E) {
    int e = blockIdx.x * blockDim.x + threadIdx.x;
    if (e < E) {
        atomicAdd(&deg[edges[2 * e + 0]], 1);
        atomicAdd(&deg[edges[2 * e + 1]], 1);
    }
}

__global__ void k_maxdeg(const int* __restrict__ deg, int* __restrict__ maxdeg, int N) {
    int i = blockIdx.x * blockDim.x + threadIdx.x;
    if (i < N) atomicMax(maxdeg, deg[i]);
}

// exclusive scan of sizes -> offsets (single block)
__global__ void k_offsets(const int* __restrict__ sizes, int* __restrict__ offsets, int B) {
    __shared__ int part[256];
    int t = threadIdx.x;
    int chunk = (B + 255) / 256;
    int s0 = t * chunk;
    int acc = 0;
    for (int j = 0; j < chunk; ++j) {
        int idx = s0 + j;
        if (idx < B) acc += sizes[idx];
    }
    part[t] = acc;
    __syncthreads();
    if (t == 0) {
        int run = 0;
        for (int i = 0; i < 256; ++i) { int tmp = part[i]; part[i] = run; run += tmp; }
    }
    __syncthreads();
    acc = part[t];
    for (int j = 0; j < chunk; ++j) {
        int idx = s0 + j;
        if (idx < B) { offsets[idx] = acc; acc += sizes[idx]; }
    }
}

// ---------------- Phase 1: TV proximal iterations ----------------

__global__ void k_copy_xy(const float* __restrict__ x, float* __restrict__ y, int N) {
    int i = blockIdx.x * blockDim.x + threadIdx.x;
    if (i < N) y[i] = x[i];
}

// y -= dT(u):  y[a] -= u, y[b] += u
__global__ void k_scatter(const int* __restrict__ edges, const float* __restrict__ u,
                          float* __restrict__ y, int E) {
    int e = blockIdx.x * blockDim.x + threadIdx.x;
    if (e < E) {
        __builtin_prefetch(&edges[2 * (e + 4096)], 0, 1);  // global_prefetch_b8
        float ue = u[e];
        atomicAdd(&y[edges[2 * e + 0]], -ue);
        atomicAdd(&y[edges[2 * e + 1]],  ue);
    }
}

__global__ void k_update_u(const int* __restrict__ edges, const float* __restrict__ y,
                           float* __restrict__ u, const int* __restrict__ maxdeg, int E) {
    int e = blockIdx.x * blockDim.x + threadIdx.x;
    if (e < E) {
        float step = 1.0f / (2.0f * (float)(*maxdeg));
        int a = edges[2 * e + 0];
        int b = edges[2 * e + 1];
        float v = u[e] + step * (y[a] - y[b]);
        v = fminf(fmaxf(v, -LAMF), LAMF);
        u[e] = v;
    }
}

// ---------------- Phase 2: per-group sparsemax ----------------
// One block per group: async-stage group data into LDS, bitonic sort (desc),
// then cumsum of the first 1536 sorted values using V_WMMA_F32_16X16X4_F32
// row-prefix matmuls (rowprefix = Z * U, U = upper-triangular ones), then
// support search for tau and the final thresholding.

__global__ __launch_bounds__(PM_TPB)
void k_sparsemax(const float* __restrict__ y, const int* __restrict__ sizes,
                 const int* __restrict__ offsets, float* __restrict__ out) {
    __shared__ float s_sh[SORTN];       // sorted values (padded with -1e30)
    __shared__ float cs_sh[NMAX_PAD];   // cumulative sums
    __shared__ float rowTot[96];
    __shared__ float rowOff[96];
    __shared__ int   kmax_sh;

    int g   = blockIdx.x;
    int tid = threadIdx.x;
    int sz  = sizes[g];
    int off = offsets[g];

    if (tid == 0) kmax_sh = 0;

#if HAVE_ASYNC_LDS
    // Pad tail via DS stores; stream live elements global->LDS on the async
    // copy path (ASYNCcnt). Address ranges are disjoint, so the two paths
    // need no mutual ordering; both are fenced before the barrier.
    for (int i = sz + tid; i < SORTN; i += PM_TPB)
        s_sh[i] = -1.0e30f;
    for (int i = tid; i < sz; i += PM_TPB)
        __builtin_amdgcn_global_load_async_to_lds_b32(
            (gint_as1*)(float*)(y + off + i),
            (lint_as3*)(s_sh + i), 0, 0);
    asm volatile("s_wait_asynccnt 0x0" ::: "memory");
#else
    for (int i = tid; i < SORTN; i += PM_TPB)
        s_sh[i] = (i < sz) ? y[off + i] : -1.0e30f;
#endif
    __syncthreads();

    // Bitonic sort, descending.
    for (int k = 2; k <= SORTN; k <<= 1) {
        for (int j = k >> 1; j > 0; j >>= 1) {
            for (int i = tid; i < SORTN; i += PM_TPB) {
                int ixj = i ^ j;
                if (ixj > i) {
                    float a = s_sh[i];
                    float c = s_sh[ixj];
                    bool swp = ((i & k) == 0) ? (a < c) : (a > c);
                    if (swp) { s_sh[i] = c; s_sh[ixj] = a; }
                }
            }
            __syncthreads();
        }
    }

    // WMMA row-prefix: waves 0..5 each handle one 16x16 tile (256 values).
    // Z[r][c] = s[256*w + 16r + c];  D = sum_s A_s(16x4) x B_s(4x16),
    // A_s = Z[:,4s:4s+4], B_s[k][n] = (4s+k <= n).
    int wave = tid >> 5;
    int lane = tid & 31;
    if (wave < 6) {
        int M   = lane & 15;           // A row index for this lane
        int khi = (lane >> 4) << 1;    // 0 (lanes 0-15) or 2 (lanes 16-31)
        int col = lane & 15;           // B/D column index
        const float* zb = &s_sh[256 * wave];
        v8f acc = {0.f, 0.f, 0.f, 0.f, 0.f, 0.f, 0.f, 0.f};
#pragma unroll
        for (int sb = 0; sb < 4; ++sb) {
            int k0 = 4 * sb + khi;     // K indices (k0, k0+1) held by this lane
            v2f av;
            av[0] = zb[M * 16 + k0];
            av[1] = zb[M * 16 + k0 + 1];
            v2f bv;                    // upper-triangular ones, synthesized
            bv[0] = (k0     <= col) ? 1.0f : 0.0f;
            bv[1] = (k0 + 1 <= col) ? 1.0f : 0.0f;
            acc = __builtin_amdgcn_wmma_f32_16x16x4_f32(
                /*neg_a=*/false, av, /*neg_b=*/false, bv,
                /*c_mod=*/(short)0, acc, /*reuse_a=*/false, /*reuse_b=*/false);
        }
        // D layout: VGPR r -> row r (lanes 0-15) / row r+8 (lanes 16-31)
#pragma unroll
        for (int r = 0; r < 8; ++r) {
            int row = (lane < 16) ? r : (r + 8);
            cs_sh[256 * wave + row * 16 + col] = acc[r];
        }
    }
    __syncthreads();

    // Cross-row carry: 96 row totals, exclusive scan, add back.
    if (tid < 96) rowTot[tid] = cs_sh[tid * 16 + 15];
    __syncthreads();
    if (tid < 96) {
        float o = 0.0f;
        for (int j = 0; j < tid; ++j) o += rowTot[j];
        rowOff[tid] = o;
    }
    __syncthreads();
    for (int i = tid; i < NMAX_PAD; i += PM_TPB)
        if (i >= 16) cs_sh[i] += rowOff[i >> 4];
    __syncthreads();

    // Support search: largest k with 1 + k*zs[k-1] > cs[k-1].
    int lim = (sz < NMAX_PAD) ? sz : NMAX_PAD;
    for (int i = tid; i < lim; i += PM_TPB) {
        float zs = s_sh[i];
        float cs = cs_sh[i];
        if (1.0f + (float)(i + 1) * zs > cs) atomicMax(&kmax_sh, i + 1);
    }
    __syncthreads();

    int   kmax = kmax_sh;
    float tau  = (cs_sh[kmax - 1] - 1.0f) / (float)kmax;
    float scl  = (float)sz;
    for (int i = tid; i < sz; i += PM_TPB) {
        float p = y[off + i] - tau;
        out[off + i] = (p > 0.0f ? p : 0.0f) * scl;
    }
}

// ---------------- host launch ----------------

extern "C" void kernel_launch(void* const* d_in, const int* in_sizes, int n_in,
                              void* d_out, int out_size, void* d_ws, size_t ws_size,
                              hipStream_t stream) {
    const float* x     = (const float*)d_in[0];
    const int*   sizes = (const int*)d_in[1];
    const int*   edges = (const int*)d_in[2];
    float*       out   = (float*)d_out;

    int N = in_sizes[0];
    int B = in_sizes[1];
    int E = in_sizes[2] / 2;

    char* ws = (char*)d_ws;
    float* y       = (float*)(ws);
    float* u       = (float*)(ws + (size_t)N * 4);
    int*   deg     = (int*)  (ws + (size_t)(N + E) * 4);
    int*   maxdeg  = (int*)  (ws + (size_t)(2 * N + E) * 4);
    int*   offsets = (int*)  (ws + (size_t)(2 * N + E + 1) * 4);

    const int tpb = 256;
    int gN = (N + tpb - 1) / tpb;
    int gE = (E + tpb - 1) / tpb;
    int gM = ((N > E ? N : E) + tpb - 1) / tpb;

    k_zero   <<<gM, tpb, 0, stream>>>(u, deg, maxdeg, E, N);
    k_deg    <<<gE, tpb, 0, stream>>>(edges, deg, E);
    k_maxdeg <<<gN, tpb, 0, stream>>>(deg, maxdeg, N);
    k_offsets<<<1, 256, 0, stream>>>(sizes, offsets, B);

    for (int it = 0; it < TV_ITERS; ++it) {
        k_copy_xy <<<gN, tpb, 0, stream>>>(x, y, N);
        k_scatter <<<gE, tpb, 0, stream>>>(edges, u, y, E);
        k_update_u<<<gE, tpb, 0, stream>>>(edges, y, u, maxdeg, E);
    }
    // final y = x - dT(u)
    k_copy_xy<<<gN, tpb, 0, stream>>>(x, y, N);
    k_scatter<<<gE, tpb, 0, stream>>>(edges, u, y, E);

    k_sparsemax<<<B, PM_TPB, 0, stream>>>(y, sizes, offsets, out);
}